// Verblizer_model_55456617726412
// MI455X (gfx1250) — compile-verified
//
#include <hip/hip_runtime.h>
#include <math.h>

// MI455X / gfx1250 (wave32, WMMA).
//
// Roofline: the workload moves ~230 MB (x=160MB dominates) and does ~120 MFLOP
// -> pure HBM-bound, ~10us at 23.3 TB/s. We therefore spend "free" matrix
// throughput to avoid per-token W gathers: compute the dense product of each
// 16-token h-tile against ALL 20 experts with V_WMMA_F32_16X16X4_F32
// (Y48 = A(16x4) x B(4x48) + bias), then each token selects its expert's two
// columns through LDS and applies a 2-way softmax.

typedef float v2f __attribute__((ext_vector_type(2)));
typedef float v8f __attribute__((ext_vector_type(8)));

#define NE 20  // experts / alphabet size

__global__ __launch_bounds__(256) void verblizer_wmma_kernel(
    const float* __restrict__ x,   // [S,20]
    const float* __restrict__ h,   // [S,2]
    const float* __restrict__ W,   // [20,2,2]  W[e][o][i]
    const float* __restrict__ bb,  // [20,2]
    float* __restrict__ y,         // [S,2]
    int S)
{
  // per-wave staging: Y48 tile, layout lds[col][token] -> 48*32 floats = 6 KB/wave
  __shared__ __align__(16) float ylds[8][48 * 32];
  const int lane = threadIdx.x & 31;
  const int wid  = threadIdx.x >> 5;
  const int gw   = blockIdx.x * (blockDim.x >> 5) + wid;
  const int nw   = gridDim.x * (blockDim.x >> 5);
  const int ntiles = (S + 31) >> 5;

  // ---- Precompute B tiles + bias (wave-invariant, lives in VGPRs) ----
  // B for V_WMMA_F32_16X16X4_F32 is 4x16 in 2 VGPRs:
  //   V0: lanes0-15 = row K=0, lanes16-31 = row K=2 (zero pad)
  //   V1: lanes0-15 = row K=1, lanes16-31 = row K=3 (zero pad)
  // Packed weight matrix: Wcol[k, 2e+o] = W[e][o][k]; cols 40..47 zero.
  v2f Bt[3];
  float Cb[3];
  const int n = lane & 15;
  #pragma unroll
  for (int j = 0; j < 3; ++j) {
    int c = j * 16 + n;
    float w0 = 0.f, w1 = 0.f, cb = 0.f;
    if (c < 2 * NE) {
      int e = c >> 1, o = c & 1;
      w0 = W[e * 4 + o * 2 + 0];
      w1 = W[e * 4 + o * 2 + 1];
      cb = bb[e * 2 + o];
    }
    Bt[j].x = (lane < 16) ? w0 : 0.f;
    Bt[j].y = (lane < 16) ? w1 : 0.f;
    Cb[j] = cb;  // bias broadcast over all 16 rows of the C tile
  }

  float* lds = &ylds[wid][0];

  for (int tile = gw; tile < ntiles; tile += nw) {
    int t = (tile << 5) + lane;       // this lane's token
    bool valid = (t < S);

    // ---- coalesced loads: h (float2) and x row (5 x float4, 80B rows are
    //      16B aligned), first-occurrence argmax over 20 channels ----
    float h0 = 0.f, h1 = 0.f;
    int e = 0;
    if (valid) {
      float2 hv = *(const float2*)(h + 2 * (size_t)t);
      h0 = hv.x; h1 = hv.y;
      const float4* xr = (const float4*)(x + 20 * (size_t)t);
      float4 v0 = xr[0];
      float best = v0.x; e = 0;
      if (v0.y > best) { best = v0.y; e = 1; }
      if (v0.z > best) { best = v0.z; e = 2; }
      if (v0.w > best) { best = v0.w; e = 3; }
      #pragma unroll
      for (int q = 1; q < 5; ++q) {
        float4 v = xr[q];
        if (v.x > best) { best = v.x; e = 4 * q + 0; }
        if (v.y > best) { best = v.y; e = 4 * q + 1; }
        if (v.z > best) { best = v.z; e = 4 * q + 2; }
        if (v.w > best) { best = v.w; e = 4 * q + 3; }
      }
    }

    // hide latency of the next grid-stride tile; locality=3 -> near (WGP) scope
    {
      int tn = t + (nw << 5);
      if (tn < S) __builtin_prefetch(x + 20 * (size_t)tn, 0, 3);
    }

    // ---- two 16-token WMMA groups; EXEC is full here (reconverged) ----
    #pragma unroll
    for (int g = 0; g < 2; ++g) {
      // A (16x4): lanes0-15 carry h of tokens g*16..g*16+15 in K0/K1,
      // lanes16-31 are the K2/K3 zero pad.
      float ah0 = (g == 0) ? h0 : __shfl(h0, (lane + 16) & 31, 32);
      float ah1 = (g == 0) ? h1 : __shfl(h1, (lane + 16) & 31, 32);
      v2f a;
      a.x = (lane < 16) ? ah0 : 0.f;
      a.y = (lane < 16) ? ah1 : 0.f;
      #pragma unroll
      for (int j = 0; j < 3; ++j) {
        v8f c;
        #pragma unroll
        for (int r = 0; r < 8; ++r) c[r] = Cb[j];
        v8f d = __builtin_amdgcn_wmma_f32_16x16x4_f32(
            false, a, false, Bt[j], (short)0, c, false, false);
        // D layout: VGPR r, lane l -> (M = r + 8*(l>=16), N = l&15).
        // Store to lds[col][token]: col = 16j + (l&15),
        // token = 16g + 8*(l>>4) + r  -> 8 contiguous floats per lane.
        int base = (j * 16 + (lane & 15)) * 32 + g * 16 + ((lane >> 4) << 3);
        *(float4*)(lds + base)     = make_float4(d[0], d[1], d[2], d[3]);
        *(float4*)(lds + base + 4) = make_float4(d[4], d[5], d[6], d[7]);
      }
    }
    // cross-lane LDS RAW within the wave
    asm volatile("s_wait_dscnt 0" ::: "memory");

    // ---- per-token expert-column select + 2-way softmax + store ----
    if (valid) {
      float y0 = lds[(2 * e + 0) * 32 + lane];
      float y1 = lds[(2 * e + 1) * 32 + lane];
      float mx = fmaxf(y0, y1);
      float e0 = __expf(y0 - mx);
      float e1 = __expf(y1 - mx);
      // softmax normalization: v_rcp_f32 (<=1 ulp) instead of the ~12-op
      // IEEE-exact div_scale/div_fmas sequence
      float inv = __builtin_amdgcn_rcpf(e0 + e1);
      *(float2*)(y + 2 * (size_t)t) = make_float2(e0 * inv, e1 * inv);
    }
  }
}

// y2[perm] where perm = [pu_index..., non-selected in original order...].
// pu_index is sorted+unique, so one lower_bound gives both cases:
//   lb = #selected < i ;  selected -> pos = lb ; else pos = P + (i - lb).
// pu_index is 4 MB -> L2-resident (192 MB L2), 20-step search is cheap.
__global__ __launch_bounds__(256) void pu_permute_kernel(
    const float* __restrict__ y, const int* __restrict__ pu,
    float* __restrict__ y2, int S, int P)
{
  int stride = gridDim.x * blockDim.x;
  for (int i = blockIdx.x * blockDim.x + threadIdx.x; i < S; i += stride) {
    int lo = 0, hi = P;
    while (lo < hi) {
      int mid = (lo + hi) >> 1;
      if (pu[mid] < i) lo = mid + 1; else hi = mid;
    }
    int pos = (lo < P && pu[lo] == i) ? lo : (P + i - lo);
    float2 v = *(const float2*)(y + 2 * (size_t)i);
    *(float2*)(y2 + 2 * (size_t)pos) = v;
  }
}

extern "C" void kernel_launch(void* const* d_in, const int* in_sizes, int n_in,
                              void* d_out, int out_size, void* d_ws, size_t ws_size,
                              hipStream_t stream) {
  const float* x  = (const float*)d_in[0];   // [1,S,20]
  const float* h  = (const float*)d_in[1];   // [S,2]
  const float* W  = (const float*)d_in[2];   // [20,2,2]
  const float* b  = (const float*)d_in[3];   // [20,2]
  const int*   pu = (const int*)d_in[4];     // [P]

  const int S = in_sizes[1] / 2;
  const int P = in_sizes[4];

  float* y  = (float*)d_out;                 // output1: [S,2]
  float* y2 = y + (size_t)2 * S;             // output2: [S,2]

  // 1024 WGs x 8 waves = 8192 waves over 62500 32-token tiles (~8 tiles/wave)
  verblizer_wmma_kernel<<<1024, 256, 0, stream>>>(x, h, W, b, y, S);
  // scatter permutation (depends on y; same stream serializes)
  pu_permute_kernel<<<2048, 256, 0, stream>>>(y, pu, y2, S, P);
}